// GCN_11081015624039
// MI455X (gfx1250) — compile-verified
//
#include <hip/hip_runtime.h>

#define N_NODES 100000
#define E_EDGES 1600000
#define F_IN    512
#define H1      64
#define H2      32
#define NCLS    10
#define EPSBN   1e-5f
#define SLOPE   0.01f

typedef float v2f __attribute__((ext_vector_type(2)));
typedef float v8f __attribute__((ext_vector_type(8)));
typedef unsigned int u32x4 __attribute__((ext_vector_type(4)));
typedef int i32x4 __attribute__((ext_vector_type(4)));
typedef int i32x8 __attribute__((ext_vector_type(8)));

#if __has_builtin(__builtin_amdgcn_tensor_load_to_lds) && \
    __has_builtin(__builtin_amdgcn_s_wait_tensorcnt)
#define HAVE_TDM 1
#endif

__device__ __forceinline__ v8f wmma4(v2f a, v2f b, v8f c) {
  // V_WMMA_F32_16X16X4_F32 : exact fp32 D = A(16x4) * B(4x16) + C
  return __builtin_amdgcn_wmma_f32_16x16x4_f32(false, a, false, b, (short)0, c,
                                               false, false);
}

__device__ __forceinline__ float lrelu(float v) { return v > 0.f ? v : SLOPE * v; }

// Stage nElems f32 from global into LDS. TDM (async tensor DMA) when available,
// issued by wave 0 only (TENSORcnt is per-wave); cooperative copy otherwise.
// Caller must __syncthreads() after.
__device__ __forceinline__ void stage_lds(float* lds_dst, const float* gsrc,
                                          unsigned nElems) {
#ifdef HAVE_TDM
  if (threadIdx.x < 32) {
    unsigned long long ga = (unsigned long long)gsrc;
    unsigned lo32 = (unsigned)(unsigned long long)(void*)lds_dst; // LDS byte offset
    u32x4 g0;
    g0[0] = 1u;                                   // count=1, user descriptor
    g0[1] = lo32;                                 // lds_addr
    g0[2] = (unsigned)ga;                         // global_addr[31:0]
    g0[3] = (unsigned)(ga >> 32) | (2u << 30);    // global_addr[56:32] | type=2
    i32x8 g1;
    g1[0] = (int)(2u << 16);                      // data_size=4B, mask=0, no flags
    g1[1] = (int)((nElems & 0xFFFFu) << 16);      // tensor_dim0[15:0]
    g1[2] = (int)(((nElems >> 16) & 0xFFFFu) | (1u << 16)); // dim0 hi | tensor_dim1=1
    g1[3] = (int)((nElems & 0xFFFFu) << 16);      // tile_dim0 = nElems
    g1[4] = 1;                                    // tile_dim1=1, tile_dim2=0
    g1[5] = (int)nElems;                          // tensor_dim0_stride
    g1[6] = 0; g1[7] = 0;
    i32x4 z4 = {0, 0, 0, 0};
#if __has_include(<hip/amd_detail/amd_gfx1250_TDM.h>)
    i32x8 z8 = {0, 0, 0, 0, 0, 0, 0, 0};          // clang-23/therock 6-arg form
    __builtin_amdgcn_tensor_load_to_lds(g0, g1, z4, z4, z8, 0);
#else
    __builtin_amdgcn_tensor_load_to_lds(g0, g1, z4, z4, 0); // ROCm 7.2 5-arg form
#endif
    __builtin_amdgcn_s_wait_tensorcnt(0);
  }
#else
  for (unsigned i = threadIdx.x; i < nElems; i += 256) lds_dst[i] = gsrc[i];
#endif
}

// ---------------------------------------------------------------- init
__global__ void k_init(float* __restrict__ deg, float* __restrict__ stats) {
  int i = blockIdx.x * 256 + threadIdx.x;
  if (i < N_NODES) deg[i] = 1.0f;      // self-loop weight
  if (i < 192) stats[i] = 0.0f;        // sum1[64] sq1[64] sum2[32] sq2[32]
}

// ---------------- pack W1 -> pair-interleaved [k/2][col][2] for b64 fragments
__global__ void k_pack(const float* __restrict__ W1, float* __restrict__ W1p) {
  int idx = blockIdx.x * 256 + threadIdx.x;          // F_IN*H1 = 32768 exactly
  int k = idx >> 6, c = idx & 63;
  W1p[(((k >> 1) * H1) + c) * 2 + (k & 1)] = W1[idx];
}

// ------------- GEMM1: h = lrelu(x@W1+b1) + BN1 stats. 32 rows/wave, W1 in LDS.
__global__ __launch_bounds__(256) void k_gemm1(
    const float* __restrict__ x, const float* __restrict__ W1p,
    const float* __restrict__ b1, float* __restrict__ h,
    float* __restrict__ sum1, float* __restrict__ sq1) {
  __shared__ float sW[(F_IN / 2) * H1 * 2];          // 32768 f32 = 128 KB
  __shared__ float redS[H1], redQ[H1];
  if (threadIdx.x < H1) { redS[threadIdx.x] = 0.f; redQ[threadIdx.x] = 0.f; }
  stage_lds(sW, W1p, (F_IN / 2) * H1 * 2);           // TDM -> LDS
  __syncthreads();

  const int lane = threadIdx.x & 31;
  const int wave = threadIdx.x >> 5;
  const int lo = lane & 15, hi = lane >> 4;
  const int rowBase = (blockIdx.x * 8 + wave) * 32;  // 32 rows per wave

  int r0 = rowBase + lo;      if (r0 >= N_NODES) r0 = N_NODES - 1;
  int r1 = rowBase + 16 + lo; if (r1 >= N_NODES) r1 = N_NODES - 1;
  const float* __restrict__ xr0 = x + (size_t)r0 * F_IN + 2 * hi;
  const float* __restrict__ xr1 = x + (size_t)r1 * F_IN + 2 * hi;

  v8f acc[2][4] = {};                                // 2 row-groups x 64 cols
  for (int k = 0; k < F_IN; k += 4) {
    v2f a0 = __builtin_nontemporal_load((const v2f*)(xr0 + k));
    v2f a1 = __builtin_nontemporal_load((const v2f*)(xr1 + k));
    const float* wk = sW + ((k >> 1) + hi) * (H1 * 2);
#pragma unroll
    for (int t = 0; t < 4; ++t) {
      v2f b = *(const v2f*)(wk + (t * 16 + lo) * 2); // ds_load_b64
      acc[0][t] = wmma4(a0, b, acc[0][t]);
      acc[1][t] = wmma4(a1, b, acc[1][t]);
    }
  }

  float s[4] = {0.f, 0.f, 0.f, 0.f}, q[4] = {0.f, 0.f, 0.f, 0.f};
#pragma unroll
  for (int g = 0; g < 2; ++g) {
#pragma unroll
    for (int t = 0; t < 4; ++t) {
      const int col = t * 16 + lo;
      const float bias = b1[col];
#pragma unroll
      for (int r = 0; r < 8; ++r) {
        int row = rowBase + 16 * g + r + 8 * hi;     // C layout: VGPR r -> M=r+8*hi
        float v = lrelu(acc[g][t][r] + bias);
        if (row < N_NODES) {
          h[(size_t)row * H1 + col] = v;
          s[t] += v; q[t] += v * v;
        }
      }
    }
  }
#pragma unroll
  for (int t = 0; t < 4; ++t) {
    const int col = t * 16 + lo;
    atomicAdd(&redS[col], s[t]);                     // ds_add_f32
    atomicAdd(&redQ[col], q[t]);
  }
  __syncthreads();
  if (threadIdx.x < H1) {
    atomicAdd(&sum1[threadIdx.x], redS[threadIdx.x]);
    atomicAdd(&sq1[threadIdx.x],  redQ[threadIdx.x]);
  }
}

// --- BN1 params + fold BN1 into GCN weight (pair-interleaved Wc', and c0)
__global__ void k_bn1(const float* __restrict__ sum1, const float* __restrict__ sq1,
                      const float* __restrict__ g1, const float* __restrict__ be1,
                      const float* __restrict__ Wc, float* __restrict__ scale1,
                      float* __restrict__ shift1, float* __restrict__ WcPp,
                      float* __restrict__ c0) {
  __shared__ float sh[H1];
  const int k = threadIdx.x;                          // 64 threads
  const float invN = 1.0f / (float)N_NODES;
  float mean = sum1[k] * invN;
  float var  = sq1[k] * invN - mean * mean;           // biased variance
  float sc = g1[k] * rsqrtf(var + EPSBN);
  float sf = be1[k] - mean * sc;
  scale1[k] = sc; shift1[k] = sf; sh[k] = sf;
  __syncthreads();
  for (int j = 0; j < H2; ++j)
    WcPp[(((k >> 1) * H2) + j) * 2 + (k & 1)] = sc * Wc[k * H2 + j];
  if (k < H2) {
    float a = 0.f;
    for (int kk = 0; kk < H1; ++kk) a += sh[kk] * Wc[kk * H2 + k];
    c0[k] = a;                                        // shift1 @ Wc
  }
}

// ------------- GEMM2: xl = h_bn @ Wc = h @ Wc' + c0. 32 rows/wave, Wc' in LDS.
__global__ __launch_bounds__(256) void k_gemm2(
    const float* __restrict__ h, const float* __restrict__ WcPp,
    const float* __restrict__ c0, float* __restrict__ xl) {
  __shared__ float sW[(H1 / 2) * H2 * 2];             // 2048 f32 = 8 KB
  stage_lds(sW, WcPp, (H1 / 2) * H2 * 2);             // TDM -> LDS
  __syncthreads();

  const int lane = threadIdx.x & 31;
  const int wave = threadIdx.x >> 5;
  const int lo = lane & 15, hi = lane >> 4;
  const int rowBase = (blockIdx.x * 8 + wave) * 32;

  int r0 = rowBase + lo;      if (r0 >= N_NODES) r0 = N_NODES - 1;
  int r1 = rowBase + 16 + lo; if (r1 >= N_NODES) r1 = N_NODES - 1;
  const float* __restrict__ hr0 = h + (size_t)r0 * H1 + 2 * hi;
  const float* __restrict__ hr1 = h + (size_t)r1 * H1 + 2 * hi;

  v8f acc[2][2] = {};
  for (int k = 0; k < H1; k += 4) {
    v2f a0 = *(const v2f*)(hr0 + k);
    v2f a1 = *(const v2f*)(hr1 + k);
    const float* wk = sW + ((k >> 1) + hi) * (H2 * 2);
#pragma unroll
    for (int t = 0; t < 2; ++t) {
      v2f b = *(const v2f*)(wk + (t * 16 + lo) * 2);  // ds_load_b64
      acc[0][t] = wmma4(a0, b, acc[0][t]);
      acc[1][t] = wmma4(a1, b, acc[1][t]);
    }
  }
#pragma unroll
  for (int g = 0; g < 2; ++g) {
#pragma unroll
    for (int t = 0; t < 2; ++t) {
      const int col = t * 16 + lo;
      const float cc = c0[col];
#pragma unroll
      for (int r = 0; r < 8; ++r) {
        int row = rowBase + 16 * g + r + 8 * hi;
        if (row < N_NODES) xl[(size_t)row * H2 + col] = acc[g][t][r] + cc;
      }
    }
  }
}

// ---------------------------------------------------------------- degrees
__global__ void k_deg(const int* __restrict__ col, const float* __restrict__ w,
                      float* __restrict__ deg) {
  int e = blockIdx.x * 256 + threadIdx.x;
  if (e < E_EDGES) atomicAdd(&deg[col[e]], w[e]);
}

__global__ void k_dinv(const float* __restrict__ deg, float* __restrict__ dinv) {
  int i = blockIdx.x * 256 + threadIdx.x;
  if (i < N_NODES) { float d = deg[i]; dinv[i] = d > 0.f ? rsqrtf(d) : 0.f; }
}

// ------------------------------ self-loop term doubles as accumulator init
__global__ void k_self(const float* __restrict__ xl, const float* __restrict__ dinv,
                       float* __restrict__ acc) {
  int idx = blockIdx.x * 256 + threadIdx.x;           // N*32 exactly
  int i = idx >> 5;
  float di = dinv[i];
  acc[idx] = xl[idx] * di * di;
}

// ---------------------------------------------------------------- edge scatter
__global__ void k_scatter(const int* __restrict__ row, const int* __restrict__ col,
                          const float* __restrict__ w, const float* __restrict__ dinv,
                          const float* __restrict__ xl, float* __restrict__ acc) {
  int t = blockIdx.x * 256 + threadIdx.x;             // 4*E exactly
  int e = t >> 2, part = t & 3;                       // 4 threads/edge, 8 feats each
  int r = row[e], c = col[e];
  float norm = dinv[r] * w[e] * dinv[c];
  const float* src = xl + (size_t)r * H2 + part * 8;
  float* dst = acc + (size_t)c * H2 + part * 8;
  float4 v0 = *(const float4*)(src);
  float4 v1 = *(const float4*)(src + 4);
  atomicAdd(dst + 0, v0.x * norm); atomicAdd(dst + 1, v0.y * norm);
  atomicAdd(dst + 2, v0.z * norm); atomicAdd(dst + 3, v0.w * norm);
  atomicAdd(dst + 4, v1.x * norm); atomicAdd(dst + 5, v1.y * norm);
  atomicAdd(dst + 6, v1.z * norm); atomicAdd(dst + 7, v1.w * norm);
}

// ----------------------- h2 = lrelu(acc + bc) in place, + BN2 stats
__global__ __launch_bounds__(256) void k_post2(float* __restrict__ acc,
                                               const float* __restrict__ bc,
                                               float* __restrict__ sum2,
                                               float* __restrict__ sq2) {
  __shared__ float rS[H2], rQ[H2];
  if (threadIdx.x < H2) { rS[threadIdx.x] = 0.f; rQ[threadIdx.x] = 0.f; }
  __syncthreads();
  int idx = blockIdx.x * 256 + threadIdx.x;           // N*32 exactly
  int j = idx & 31;
  float v = lrelu(acc[idx] + bc[j]);
  acc[idx] = v;
  atomicAdd(&rS[j], v);
  atomicAdd(&rQ[j], v * v);
  __syncthreads();
  if (threadIdx.x < H2) {
    atomicAdd(&sum2[threadIdx.x], rS[threadIdx.x]);
    atomicAdd(&sq2[threadIdx.x],  rQ[threadIdx.x]);
  }
}

__global__ void k_bn2(const float* __restrict__ sum2, const float* __restrict__ sq2,
                      const float* __restrict__ g2, const float* __restrict__ be2,
                      float* __restrict__ scale2, float* __restrict__ shift2) {
  const int k = threadIdx.x;                          // 32 threads
  const float invN = 1.0f / (float)N_NODES;
  float mean = sum2[k] * invN;
  float var  = sq2[k] * invN - mean * mean;
  float sc = g2[k] * rsqrtf(var + EPSBN);
  scale2[k] = sc; shift2[k] = be2[k] - mean * sc;
}

// ------------------------- classifier: cat@W2 + b2, log_softmax
__global__ __launch_bounds__(256) void k_final(
    const float* __restrict__ h, const float* __restrict__ h2,
    const float* __restrict__ scale1, const float* __restrict__ shift1,
    const float* __restrict__ scale2, const float* __restrict__ shift2,
    const float* __restrict__ W2, const float* __restrict__ b2,
    float* __restrict__ out) {
  __shared__ float sW2[(H1 + H2) * NCLS];
  __shared__ float sB2[NCLS], sS1[H1], sF1[H1], sS2[H2], sF2[H2];
  for (int i = threadIdx.x; i < (H1 + H2) * NCLS; i += 256) sW2[i] = W2[i];
  if (threadIdx.x < NCLS) sB2[threadIdx.x] = b2[threadIdx.x];
  if (threadIdx.x < H1) { sS1[threadIdx.x] = scale1[threadIdx.x];
                          sF1[threadIdx.x] = shift1[threadIdx.x]; }
  if (threadIdx.x < H2) { sS2[threadIdx.x] = scale2[threadIdx.x];
                          sF2[threadIdx.x] = shift2[threadIdx.x]; }
  __syncthreads();
  int i = blockIdx.x * 256 + threadIdx.x;
  if (i >= N_NODES) return;

  float lg[NCLS];
#pragma unroll
  for (int c = 0; c < NCLS; ++c) lg[c] = sB2[c];
  const float* hr = h + (size_t)i * H1;
#pragma unroll 4
  for (int j = 0; j < H1; ++j) {
    float v = hr[j] * sS1[j] + sF1[j];
#pragma unroll
    for (int c = 0; c < NCLS; ++c) lg[c] += v * sW2[j * NCLS + c];
  }
  const float* h2r = h2 + (size_t)i * H2;
#pragma unroll 4
  for (int j = 0; j < H2; ++j) {
    float v = h2r[j] * sS2[j] + sF2[j];
#pragma unroll
    for (int c = 0; c < NCLS; ++c) lg[c] += v * sW2[(H1 + j) * NCLS + c];
  }
  float m = lg[0];
#pragma unroll
  for (int c = 1; c < NCLS; ++c) m = fmaxf(m, lg[c]);
  float s = 0.f;
#pragma unroll
  for (int c = 0; c < NCLS; ++c) s += __expf(lg[c] - m);
  float lse = __logf(s);
  float* o = out + (size_t)i * NCLS;
#pragma unroll
  for (int c = 0; c < NCLS; ++c) o[c] = lg[c] - m - lse;
}

// ---------------------------------------------------------------- launch
extern "C" void kernel_launch(void* const* d_in, const int* in_sizes, int n_in,
                              void* d_out, int out_size, void* d_ws, size_t ws_size,
                              hipStream_t stream) {
  const float* x   = (const float*)d_in[0];
  const int*   ei  = (const int*)  d_in[1];   // [2,E] flat: rows then cols
  const float* ew  = (const float*)d_in[2];
  const float* W1  = (const float*)d_in[3];
  const float* b1  = (const float*)d_in[4];
  const float* g1  = (const float*)d_in[5];
  const float* be1 = (const float*)d_in[6];
  const float* Wc  = (const float*)d_in[7];
  const float* bc  = (const float*)d_in[8];
  const float* g2  = (const float*)d_in[9];
  const float* be2 = (const float*)d_in[10];
  const float* W2  = (const float*)d_in[11];
  const float* b2  = (const float*)d_in[12];
  float* out = (float*)d_out;

  float* ws     = (float*)d_ws;
  float* h      = ws;                                  // N*64
  float* xl     = h   + (size_t)N_NODES * H1;          // N*32
  float* acc    = xl  + (size_t)N_NODES * H2;          // N*32 (h2 in place)
  float* deg    = acc + (size_t)N_NODES * H2;          // N
  float* dinv   = deg + N_NODES;                       // N
  float* stats  = dinv + N_NODES;                      // 192
  float* sum1 = stats, *sq1 = stats + 64, *sum2 = stats + 128, *sq2 = stats + 160;
  float* scale1 = stats + 192;                         // 64
  float* shift1 = scale1 + 64;                         // 64
  float* scale2 = shift1 + 64;                         // 32
  float* shift2 = scale2 + 32;                         // 32
  float* WcPp   = shift2 + 32;                         // 64*32 pair-interleaved
  float* c0     = WcPp + H1 * H2;                      // 32
  float* W1p    = c0 + 32;                             // 512*64 pair-interleaved

  const int gemmBlocks = (N_NODES / 32 + 7) / 8;       // 391 (32 rows/wave, 8 waves)

  k_init   <<<(N_NODES + 255) / 256, 256, 0, stream>>>(deg, stats);
  k_pack   <<<(F_IN * H1) / 256, 256, 0, stream>>>(W1, W1p);
  k_gemm1  <<<gemmBlocks, 256, 0, stream>>>(x, W1p, b1, h, sum1, sq1);
  k_bn1    <<<1, H1, 0, stream>>>(sum1, sq1, g1, be1, Wc, scale1, shift1, WcPp, c0);
  k_gemm2  <<<gemmBlocks, 256, 0, stream>>>(h, WcPp, c0, xl);
  k_deg    <<<E_EDGES / 256, 256, 0, stream>>>(ei + E_EDGES, ew, deg);
  k_dinv   <<<(N_NODES + 255) / 256, 256, 0, stream>>>(deg, dinv);
  k_self   <<<(N_NODES * H2) / 256, 256, 0, stream>>>(xl, dinv, acc);
  k_scatter<<<(4 * E_EDGES) / 256, 256, 0, stream>>>(ei, ei + E_EDGES, ew, dinv, xl, acc);
  k_post2  <<<(N_NODES * H2) / 256, 256, 0, stream>>>(acc, bc, sum2, sq2);
  k_bn2    <<<1, H2, 0, stream>>>(sum2, sq2, g2, be2, scale2, shift2);
  k_final  <<<(N_NODES + 255) / 256, 256, 0, stream>>>(h, acc, scale1, shift1,
                                                       scale2, shift2, W2, b2, out);
}